// FeatureConditionedIncidenceMask_45921790329386
// MI455X (gfx1250) — compile-verified
//
#include <hip/hip_runtime.h>

#define N_NODES 100000
#define N_EDGES 20000
#define NNZ_T 1000000
#define DIM 128
#define HH 32
#define KKEEP 500000
#define NBINS 65536

typedef __attribute__((ext_vector_type(2))) float v2f;
typedef __attribute__((ext_vector_type(8))) float v8f;

// ---------------- workspace layout (32-bit words) ----------------
// [0,        2560000)   edge_feats (float)
// [2560000,  2580004)   offs    (int, 20001 used)
// [2580004,  2600004)   cursor  (int)
// [2600004,  2620004)   counts  (int)
// [2620004,  3620004)   idx_list(int, CSR order of nnz ids)
// [3620004,  3685540)   hist    (int)
// [3685540,  3685542)   sel     (int)
// [3685542,  3685543)   counter (int)

__global__ void zero_kernel(int* __restrict__ w, int n) {
    int i = blockIdx.x * blockDim.x + threadIdx.x;
    if (i < n) w[i] = 0;
}

__global__ void count_kernel(const int* __restrict__ E, int* __restrict__ counts) {
    int i = blockIdx.x * blockDim.x + threadIdx.x;
    if (i < NNZ_T) atomicAdd(&counts[E[i]], 1);
}

// One block of 1024 threads: exclusive scan of 20000 counts -> offs[0..20000].
__global__ void scan_kernel(const int* __restrict__ counts, int* __restrict__ offs) {
    __shared__ int part[1024];
    int t = threadIdx.x;
    int base = t * 20;
    int lc[20];
    int local = 0;
#pragma unroll
    for (int j = 0; j < 20; ++j) {
        int idx = base + j;
        lc[j] = (idx < N_EDGES) ? counts[idx] : 0;
        local += lc[j];
    }
    part[t] = local;
    __syncthreads();
    for (int off = 1; off < 1024; off <<= 1) {
        int v = (t >= off) ? part[t - off] : 0;
        __syncthreads();
        part[t] += v;
        __syncthreads();
    }
    int run = part[t] - local;                 // exclusive prefix of this chunk
#pragma unroll
    for (int j = 0; j < 20; ++j) {
        int idx = base + j;
        if (idx < N_EDGES) offs[idx] = run;
        run += lc[j];
    }
    if (t == 1023) offs[N_EDGES] = part[1023]; // == NNZ_T
}

__global__ void cursor_init_kernel(const int* __restrict__ offs, int* __restrict__ cursor) {
    int e = blockIdx.x * blockDim.x + threadIdx.x;
    if (e < N_EDGES) cursor[e] = offs[e];
}

__global__ void fill_kernel(const int* __restrict__ E, int* __restrict__ cursor,
                            int* __restrict__ idx_list) {
    int i = blockIdx.x * blockDim.x + threadIdx.x;
    if (i >= NNZ_T) return;
    int slot = atomicAdd(&cursor[E[i]], 1);
    idx_list[slot] = i;
}

// One wave per edge: lane handles 4 consecutive features; gather-sum, then mean.
__global__ void edge_mean_kernel(const float* __restrict__ x, const int* __restrict__ V,
                                 const int* __restrict__ idx_list, const int* __restrict__ offs,
                                 float* __restrict__ ef) {
    int e = (blockIdx.x * blockDim.x + threadIdx.x) >> 5;
    int lane = threadIdx.x & 31;
    if (e >= N_EDGES) return;
    int beg = offs[e], end = offs[e + 1];
    float4 acc = make_float4(0.f, 0.f, 0.f, 0.f);
    for (int m = beg; m < end; ++m) {
        int v = V[idx_list[m]];
        const float4 xv = *(const float4*)(x + (size_t)v * DIM + lane * 4);
        acc.x += xv.x; acc.y += xv.y; acc.z += xv.z; acc.w += xv.w;
    }
    float inv = 1.0f / fmaxf((float)(end - beg), 1.0f);
    float4 r = make_float4(acc.x * inv, acc.y * inv, acc.z * inv, acc.w * inv);
    *(float4*)(ef + (size_t)e * DIM + lane * 4) = r;
}

// One wave per 16-row tile: h = relu([x[V]|ef[E]] @ W1 + b1), p = sigmoid(h @ W2 + b2).
// W1 is pre-swizzled in LDS into the exact WMMA B-operand layout:
//   sB[kk*64 + t*32 + lane] = { W1[4kk+2h][t*16+n], W1[4kk+2h+1][t*16+n] }, h=lane>>4, n=lane&15
// so every B operand is one contiguous ds_load_b64 (no repacking movs).
__global__ void __launch_bounds__(256) gemm_probs_kernel(
        const float* __restrict__ x,  const float* __restrict__ W1,
        const float* __restrict__ b1, const float* __restrict__ W2,
        const float* __restrict__ b2, const int* __restrict__ V,
        const int* __restrict__ E,    const float* __restrict__ ef,
        float* __restrict__ probs) {
    __shared__ v2f sB[64 * 64];                         // 32 KB
    for (int idx = threadIdx.x; idx < 64 * 64; idx += blockDim.x) {
        int kk = idx >> 6;
        int rem = idx & 63;
        int t = rem >> 5;
        int l = rem & 31;
        int n = t * 16 + (l & 15);
        int k = kk * 4 + 2 * (l >> 4);
        v2f w;
        w[0] = W1[k * HH + n];
        w[1] = W1[(k + 1) * HH + n];
        sB[idx] = w;
    }
    __syncthreads();

    int tile = blockIdx.x * (blockDim.x >> 5) + (threadIdx.x >> 5);
    if (tile >= NNZ_T / 16) return;

    int lane = threadIdx.x & 31;
    int lidx = lane & 15;
    int half = lane >> 4;
    int kb   = half * 2;
    int row  = tile * 16 + lidx;
    const float* xrow = x  + (size_t)V[row] * DIM;
    const float* erow = ef + (size_t)E[row] * DIM;

    v8f c0 = {};                                        // N = 0..15
    v8f c1 = {};                                        // N = 16..31

#pragma unroll 8
    for (int kk = 0; kk < 32; ++kk) {                   // K = 0..127 : node features
        v2f a = *(const v2f*)(xrow + kk * 4 + kb);
        v2f b0  = sB[kk * 64 + lane];
        v2f b1v = sB[kk * 64 + 32 + lane];
        c0 = __builtin_amdgcn_wmma_f32_16x16x4_f32(false, a, false, b0,  (short)0, c0, false, false);
        c1 = __builtin_amdgcn_wmma_f32_16x16x4_f32(false, a, false, b1v, (short)0, c1, false, false);
    }
#pragma unroll 8
    for (int kk = 0; kk < 32; ++kk) {                   // K = 128..255 : edge features
        v2f a = *(const v2f*)(erow + kk * 4 + kb);
        v2f b0  = sB[(32 + kk) * 64 + lane];
        v2f b1v = sB[(32 + kk) * 64 + 32 + lane];
        c0 = __builtin_amdgcn_wmma_f32_16x16x4_f32(false, a, false, b0,  (short)0, c0, false, false);
        c1 = __builtin_amdgcn_wmma_f32_16x16x4_f32(false, a, false, b1v, (short)0, c1, false, false);
    }

    // C layout: VGPR j <-> M = j + 8*half, N = lidx (c0) / 16+lidx (c1).
    float b1a = b1[lidx], b1b = b1[16 + lidx];
    float w2a = W2[lidx], w2b = W2[16 + lidx];
    float bias2 = b2[0];
#pragma unroll
    for (int j = 0; j < 8; ++j) {
        float t = fmaxf(c0[j] + b1a, 0.0f) * w2a + fmaxf(c1[j] + b1b, 0.0f) * w2b;
        t += __shfl_xor(t, 1, 16);
        t += __shfl_xor(t, 2, 16);
        t += __shfl_xor(t, 4, 16);
        t += __shfl_xor(t, 8, 16);
        if (lidx == j) {
            float z = t + bias2;
            probs[tile * 16 + j + 8 * half] = 1.0f / (1.0f + __expf(-z));
        }
    }
}

__global__ void hist_kernel(const float* __restrict__ probs, int* __restrict__ hist) {
    int i = blockIdx.x * blockDim.x + threadIdx.x;
    if (i >= NNZ_T) return;
    int b = (int)(probs[i] * (float)NBINS);
    b = min(max(b, 0), NBINS - 1);
    atomicAdd(&hist[b], 1);
}

__global__ void select_kernel(const int* __restrict__ hist, int* __restrict__ sel) {
    if (threadIdx.x != 0 || blockIdx.x != 0) return;
    long long acc = 0;
    int tb = 0, need = 0, found = 0;
    for (int b = NBINS - 1; b >= 0; --b) {
        int h = hist[b];
        if (acc + h >= (long long)KKEEP) { tb = b; need = (int)(KKEEP - acc); found = 1; break; }
        acc += h;
    }
    if (!found) { tb = 0; need = 0; }
    sel[0] = tb;
    sel[1] = need;
}

__global__ void mask_kernel(const float* __restrict__ probs, const int* __restrict__ sel,
                            int* __restrict__ counter, float* __restrict__ soft) {
    int i = blockIdx.x * blockDim.x + threadIdx.x;
    if (i >= NNZ_T) return;
    int b = (int)(probs[i] * (float)NBINS);
    b = min(max(b, 0), NBINS - 1);
    int tb = sel[0];
    float s;
    if (b > tb) {
        s = 1.0f;
    } else if (b == tb) {
        int c = atomicAdd(counter, 1);
        s = (c < sel[1]) ? 1.0f : 0.0f;
    } else {
        s = 0.0f;
    }
    soft[i] = s;
}

// One wave per edge: mean of soft over CSR members, write 3 edge outputs.
__global__ void edge_out_kernel(const float* __restrict__ soft, const int* __restrict__ idx_list,
                                const int* __restrict__ offs, float* __restrict__ out) {
    int e = (blockIdx.x * blockDim.x + threadIdx.x) >> 5;
    int lane = threadIdx.x & 31;
    if (e >= N_EDGES) return;
    int beg = offs[e], end = offs[e + 1];
    float s = 0.0f;
    for (int m = beg + lane; m < end; m += 32) s += soft[idx_list[m]];
    s += __shfl_xor(s, 1, 32);
    s += __shfl_xor(s, 2, 32);
    s += __shfl_xor(s, 4, 32);
    s += __shfl_xor(s, 8, 32);
    s += __shfl_xor(s, 16, 32);
    if (lane == 0) {
        float es = s / fmaxf((float)(end - beg), 1.0f);
        out[e] = es;                                      // edge_probs
        out[N_EDGES + e] = es;                            // edge_soft
        out[2 * N_EDGES + e] = (es > 0.0f) ? 1.0f : 0.0f; // edge_hard
    }
}

extern "C" void kernel_launch(void* const* d_in, const int* in_sizes, int n_in,
                              void* d_out, int out_size, void* d_ws, size_t ws_size,
                              hipStream_t stream) {
    const float* x  = (const float*)d_in[0];
    const float* W1 = (const float*)d_in[1];
    const float* b1 = (const float*)d_in[2];
    const float* W2 = (const float*)d_in[3];
    const float* b2 = (const float*)d_in[4];
    const int*   V  = (const int*)d_in[5];
    const int*   E  = (const int*)d_in[6];

    float* out       = (float*)d_out;
    float* probs_out = out + 3 * N_EDGES;               // [1M]
    float* soft_out  = out + 3 * N_EDGES + NNZ_T;       // [1M]

    int*   ws_i      = (int*)d_ws;
    float* edge_ft   = (float*)d_ws;                    // 2,560,000 f32
    int*   offs      = ws_i + 2560000;                  // 20,001 int (padded to 20,004)
    int*   cursor    = ws_i + 2580004;                  // 20,000 int
    int*   counts    = ws_i + 2600004;                  // 20,000 int
    int*   idx_list  = ws_i + 2620004;                  // 1,000,000 int
    int*   hist      = ws_i + 3620004;                  // 65,536 int
    int*   sel       = ws_i + 3685540;                  // 2 int
    int*   counter   = ws_i + 3685542;                  // 1 int

    // zero: counts ; hist+sel+counter (contiguous 65539 words)
    zero_kernel<<<(N_EDGES + 255) / 256, 256, 0, stream>>>(counts, N_EDGES);
    zero_kernel<<<(NBINS + 3 + 255) / 256, 256, 0, stream>>>(hist, NBINS + 3);

    count_kernel<<<(NNZ_T + 255) / 256, 256, 0, stream>>>(E, counts);
    scan_kernel<<<1, 1024, 0, stream>>>(counts, offs);
    cursor_init_kernel<<<(N_EDGES + 255) / 256, 256, 0, stream>>>(offs, cursor);
    fill_kernel<<<(NNZ_T + 255) / 256, 256, 0, stream>>>(E, cursor, idx_list);
    edge_mean_kernel<<<(N_EDGES + 7) / 8, 256, 0, stream>>>(x, V, idx_list, offs, edge_ft);
    gemm_probs_kernel<<<(NNZ_T / 16 + 7) / 8, 256, 0, stream>>>(x, W1, b1, W2, b2, V, E,
                                                                edge_ft, probs_out);
    hist_kernel<<<(NNZ_T + 255) / 256, 256, 0, stream>>>(probs_out, hist);
    select_kernel<<<1, 32, 0, stream>>>(hist, sel);
    mask_kernel<<<(NNZ_T + 255) / 256, 256, 0, stream>>>(probs_out, sel, counter, soft_out);
    edge_out_kernel<<<(N_EDGES + 7) / 8, 256, 0, stream>>>(soft_out, idx_list, offs, out);
}